// DressedQuantumCircuit_45956150067936
// MI455X (gfx1250) — compile-verified
//
#include <hip/hip_runtime.h>
#include <math.h>

// ---------------------------------------------------------------------------
// DressedQuantumCircuit for MI455X (gfx1250, wave32)
//   Stage 1: angles = (x @ pre_w^T + pre_b) * pi/2      -- WMMA f32 16x16x4
//   Stage 2: 12-qubit real state-vector sim, state in 128 VGPRs per lane,
//            lane-bit qubits via ds_swizzle, reg-bit qubits in-register;
//            fused post-linear layer.
// ---------------------------------------------------------------------------

#define NQ    12
#define NL    6
#define NB    8192
#define DIN   512
#define NPAD  16   // qubit dim padded to WMMA tile width

typedef float v2f __attribute__((ext_vector_type(2)));
typedef float v8f __attribute__((ext_vector_type(8)));

// ============================ Stage 1: pre-GEMM =============================
// Grid: 256 blocks x 64 threads (2 waves) -> 512 tiles spread over many WGPs.
// Each wave owns one 16(batch) x 16(qubit) tile; K-loop of 128
// V_WMMA_F32_16X16X4_F32 accumulating in 8 VGPRs.
__global__ __launch_bounds__(64) void pre_gemm_wmma(
    const float* __restrict__ x,      // [NB, DIN]
    const float* __restrict__ pre_w,  // [NQ, DIN]
    const float* __restrict__ pre_b,  // [NQ]
    float* __restrict__ angles)       // [NB, NPAD]
{
  // Stage padded pre_w (rows 12..15 zero) into LDS once per workgroup.
  __shared__ float wl[NPAD * DIN];   // 32 KB
  for (int i = threadIdx.x; i < NPAD * DIN; i += 64) {
    int n = i >> 9;            // DIN = 512 = 2^9
    int k = i & (DIN - 1);
    wl[i] = (n < NQ) ? pre_w[n * DIN + k] : 0.0f;
  }

  const int lane = threadIdx.x & 31;
  const int wid  = threadIdx.x >> 5;
  const int tile = blockIdx.x * 2 + wid;     // 512 tiles total
  const int m0   = tile * 16;
  const int half = lane >> 4;                // 0 / 1
  const int l16  = lane & 15;

  // A frag (32-bit 16x4): lane<16 -> K={0,1}, lane>=16 -> K={2,3}, M = l16.
  const float* arow = x  + (size_t)(m0 + l16) * DIN + 2 * half;
  // B frag (32-bit 4x16): same K split, N = l16.  B[k][n] = pre_w[n][k].
  const float* brow = wl + l16 * DIN + 2 * half;

  // Warm GL2 for this wave's 16 x 2KB A rows while staging pre_w.
  __builtin_prefetch(arow, 0, 1);
  __builtin_prefetch(arow + 256, 0, 1);

  __syncthreads();

  v8f acc = {0.f, 0.f, 0.f, 0.f, 0.f, 0.f, 0.f, 0.f};
  #pragma unroll 8
  for (int k = 0; k < DIN; k += 4) {
    v2f a = *(const v2f*)(arow + k);
    v2f b = *(const v2f*)(brow + k);
    acc = __builtin_amdgcn_wmma_f32_16x16x4_f32(
        /*neg_a=*/false, a, /*neg_b=*/false, b,
        /*c_mod=*/(short)0, acc, /*reuse_a=*/false, /*reuse_b=*/false);
  }

  const float bias = (l16 < NQ) ? pre_b[l16] : 0.0f;
  const float sc = 1.5707963267948966f;  // pi/2
  // C/D layout: VGPR r -> M = m0 + r + 8*half, N = l16.
  #pragma unroll
  for (int r = 0; r < 8; ++r) {
    int m = m0 + r + 8 * half;
    angles[m * NPAD + l16] = (acc[r] + bias) * sc;
  }
}

// ============================ Stage 2 helpers ===============================
#define SWZ_XOR(m) ((((m) & 31) << 10) | 0x1f)   // group-of-32 xor swizzle

template <int PAT>
__device__ __forceinline__ float fswz(float v) {
  return __int_as_float(__builtin_amdgcn_ds_swizzle(__float_as_int(v), PAT));
}

// RY on a lane-bit qubit W (0..4): partner via xor-swizzle.
// new = c*a + (bit? +s : -s)*partner
template <int W>
__device__ __forceinline__ void ry_lane(float (&st)[128], int lane, float c, float s) {
  const float ssgn = ((lane >> W) & 1) ? s : -s;
  #pragma unroll
  for (int r = 0; r < 128; ++r) {
    float p = fswz<SWZ_XOR(1 << W)>(st[r]);
    st[r] = __builtin_fmaf(c, st[r], ssgn * p);
  }
}

// RY on a register-bit qubit (reg bit T, qubit 5+T): in-register pairs.
template <int T>
__device__ __forceinline__ void ry_reg(float (&st)[128], float c, float s) {
  #pragma unroll
  for (int r0 = 0; r0 < 128; ++r0) {
    if (r0 & (1 << T)) continue;           // compile-time skip
    const int r1 = r0 | (1 << T);
    float a0 = st[r0], a1 = st[r1];
    st[r0] = __builtin_fmaf(c, a0, -s * a1);
    st[r1] = __builtin_fmaf(c, a1,  s * a0);
  }
}

// CNOT ctrl=lane bit CW, tgt=lane bit CW+1 (CW 0..3).
template <int CW>
__device__ __forceinline__ void cnot_ll(float (&st)[128], int lane) {
  const bool ctrl = (lane >> CW) & 1;
  #pragma unroll
  for (int r = 0; r < 128; ++r) {
    float p = fswz<SWZ_XOR(1 << (CW + 1))>(st[r]);
    st[r] = ctrl ? p : st[r];
  }
}

// CNOT ctrl=lane bit 4, tgt=reg bit 0 (qubit 5).
__device__ __forceinline__ void cnot_lr(float (&st)[128], int lane) {
  const bool ctrl = (lane >> 4) & 1;
  #pragma unroll
  for (int r0 = 0; r0 < 128; r0 += 2) {
    float a = st[r0], b = st[r0 + 1];
    st[r0]     = ctrl ? b : a;
    st[r0 + 1] = ctrl ? a : b;
  }
}

// CNOT ctrl=reg bit CT, tgt=reg bit CT+1 (CT 0..5): pure register swap.
template <int CT>
__device__ __forceinline__ void cnot_rr(float (&st)[128]) {
  #pragma unroll
  for (int r = 0; r < 128; ++r) {
    if ((r & (1 << CT)) && !(r & (1 << (CT + 1)))) {
      const int r2 = r | (1 << (CT + 1));
      float t = st[r]; st[r] = st[r2]; st[r2] = t;
    }
  }
}

// ============================ Stage 2: simulate =============================
// One wave32 per batch row; 4096 amplitudes = 32 lanes x 128 VGPRs.
// Amplitude index i: qubit q(0..4) = lane bit q ; qubit 5+t = reg bit t.
__global__ __launch_bounds__(256) void sim_kernel(
    const float* __restrict__ angles,   // [NB, NPAD]
    const float* __restrict__ weights,  // [NL, NQ]
    const float* __restrict__ post_w,   // [1, NQ]
    const float* __restrict__ post_b,   // [1]
    float* __restrict__ out)            // [NB]
{
  const int lane = threadIdx.x & 31;
  const int wid  = threadIdx.x >> 5;
  const int b    = blockIdx.x * 8 + wid;

  // Initial product state: v0 = (c-s)/sqrt2, v1 = (c+s)/sqrt2 per qubit.
  const float inv_sqrt2 = 0.7071067811865476f;
  float v0[NQ], v1[NQ];
  #pragma unroll
  for (int q = 0; q < NQ; ++q) {
    float h = 0.5f * angles[b * NPAD + q];
    float c = cosf(h), s = sinf(h);
    v0[q] = (c - s) * inv_sqrt2;
    v1[q] = (c + s) * inv_sqrt2;
  }
  float lp = 1.0f;
  #pragma unroll
  for (int q = 0; q < 5; ++q) lp *= ((lane >> q) & 1) ? v1[q] : v0[q];

  float st[128];
  #pragma unroll
  for (int r = 0; r < 128; ++r) {
    float p = lp;
    #pragma unroll
    for (int t = 0; t < 7; ++t) p *= ((r >> t) & 1) ? v1[5 + t] : v0[5 + t];
    st[r] = p;
  }

  // Circuit layers (runtime loop to bound code size; gates fully unrolled).
  #pragma unroll 1
  for (int layer = 0; layer < NL; ++layer) {
    const float* wr = weights + layer * NQ;
    float cc[NQ], ss[NQ];
    #pragma unroll
    for (int q = 0; q < NQ; ++q) {
      float h = 0.5f * wr[q];
      cc[q] = cosf(h); ss[q] = sinf(h);
    }
    ry_lane<0>(st, lane, cc[0], ss[0]);
    ry_lane<1>(st, lane, cc[1], ss[1]);
    ry_lane<2>(st, lane, cc[2], ss[2]);
    ry_lane<3>(st, lane, cc[3], ss[3]);
    ry_lane<4>(st, lane, cc[4], ss[4]);
    ry_reg<0>(st, cc[5],  ss[5]);
    ry_reg<1>(st, cc[6],  ss[6]);
    ry_reg<2>(st, cc[7],  ss[7]);
    ry_reg<3>(st, cc[8],  ss[8]);
    ry_reg<4>(st, cc[9],  ss[9]);
    ry_reg<5>(st, cc[10], ss[10]);
    ry_reg<6>(st, cc[11], ss[11]);
    cnot_ll<0>(st, lane);
    cnot_ll<1>(st, lane);
    cnot_ll<2>(st, lane);
    cnot_ll<3>(st, lane);
    cnot_lr(st, lane);
    cnot_rr<0>(st);
    cnot_rr<1>(st);
    cnot_rr<2>(st);
    cnot_rr<3>(st);
    cnot_rr<4>(st);
    cnot_rr<5>(st);
  }

  // <Z_w> = sum_i sign_w(i) * st[i]^2
  float tot = 0.f;
  float acc[7] = {0.f, 0.f, 0.f, 0.f, 0.f, 0.f, 0.f};
  #pragma unroll
  for (int r = 0; r < 128; ++r) {
    float sq = st[r] * st[r];
    tot += sq;
    #pragma unroll
    for (int t = 0; t < 7; ++t) acc[t] += ((r >> t) & 1) ? -sq : sq;
  }
  float z[NQ];
  #pragma unroll
  for (int w = 0; w < 5; ++w) z[w] = ((lane >> w) & 1) ? -tot : tot;
  #pragma unroll
  for (int w = 5; w < NQ; ++w) z[w] = acc[w - 5];

  // Wave32 xor-swizzle reduction (all lanes end with the full sum).
  #pragma unroll
  for (int w = 0; w < NQ; ++w) {
    float v = z[w];
    v += fswz<SWZ_XOR(16)>(v);
    v += fswz<SWZ_XOR(8)>(v);
    v += fswz<SWZ_XOR(4)>(v);
    v += fswz<SWZ_XOR(2)>(v);
    v += fswz<SWZ_XOR(1)>(v);
    z[w] = v;
  }

  if (lane == 0) {
    float r = post_b[0];
    #pragma unroll
    for (int w = 0; w < NQ; ++w) r = __builtin_fmaf(post_w[w], z[w], r);
    out[b] = r;
  }
}

// ================================ launcher ==================================
extern "C" void kernel_launch(void* const* d_in, const int* in_sizes, int n_in,
                              void* d_out, int out_size, void* d_ws, size_t ws_size,
                              hipStream_t stream) {
  (void)in_sizes; (void)n_in; (void)out_size; (void)ws_size;
  const float* x       = (const float*)d_in[0];  // [8192, 512]
  const float* pre_w   = (const float*)d_in[1];  // [12, 512]
  const float* pre_b   = (const float*)d_in[2];  // [12]
  const float* weights = (const float*)d_in[3];  // [6, 12]
  const float* post_w  = (const float*)d_in[4];  // [1, 12]
  const float* post_b  = (const float*)d_in[5];  // [1]
  float* out    = (float*)d_out;                 // [8192]
  float* angles = (float*)d_ws;                  // [8192, 16] = 512 KB scratch

  // Stage 1: 512 tiles of 16 rows, 2 waves per block -> 256 blocks.
  pre_gemm_wmma<<<256, 64, 0, stream>>>(x, pre_w, pre_b, angles);
  // Stage 2: one wave per batch row, 8 waves per block -> 1024 blocks.
  sim_kernel<<<1024, 256, 0, stream>>>(angles, weights, post_w, post_b, out);
}